// MultiHeadCrossAttentionModule_944892805456
// MI455X (gfx1250) — compile-verified
//
#include <hip/hip_runtime.h>

// MI455X / gfx1250, wave32. Fused pair-bias cross-attention.
// One workgroup (256 threads = 8 waves) per query token (b,i). Grid = B*L = 512.
//
// Algebra: K/V projections folded through the attention so the (B,L,L,D)
// K/V tensors are never materialized:
//   scores[h,j] ~ (pair_row @ qk^T)   with qk[h,c] = (1/8) * sum_d Q[h,d]*Wk[c,h*64+d]
//   context[h,:] = (w[h,:] @ pair_row) @ Wv[:, h*64:...] + bv   (sum_j w = 1)
// bk drops out (uniform shift over j -> softmax invariant).
// Both GEMMs use V_WMMA_F32_16X16X4_F32 (exact fp32).

typedef __attribute__((ext_vector_type(2))) float v2f;
typedef __attribute__((ext_vector_type(8))) float v8f;

#define Bn   2
#define Ln   256
#define CBn  64
#define Dn   512
#define Hn   8
#define HDn  64
#define PSTR 68   // padded LDS stride for pair tile (bank-conflict free A-frags)
#define SSTR 17   // padded LDS stride for score tile

__global__ __launch_bounds__(256)
void pair_attn_fused(const float* __restrict__ x,
                     const float* __restrict__ pair,
                     const float* __restrict__ Wq, const float* __restrict__ bq,
                     const float* __restrict__ Wk,
                     const float* __restrict__ Wv, const float* __restrict__ bv,
                     const float* __restrict__ Wo, const float* __restrict__ bo,
                     const float* __restrict__ gamma, const float* __restrict__ beta,
                     float* __restrict__ out)
{
    __shared__ float P[Ln * PSTR];     // pair row tile (256 x 64, stride 68)  68.0 KB
    __shared__ float xrow[Dn];         // residual / Q input                    2 KB
    __shared__ float qrow[Dn];         // Q[b,i,:]                              2 KB
    __shared__ float qkT[CBn * 16];    // qk^T: [c][h], h padded to 16          4 KB
    __shared__ float s[Ln * SSTR];     // scores -> exp weights [j][h16]       17 KB
    __shared__ float wp[16 * CBn];     // softmax(w) @ pair  [h16][c]           4 KB
    __shared__ float ctx[Dn];          // attention context row                 2 KB
    __shared__ float rsum[16];         // 1/sum(exp) per head
    __shared__ float redS[8], redQ[8]; // LN reduction scratch

    const int tid  = threadIdx.x;
    const int wave = tid >> 5;
    const int lane = tid & 31;
    const int half = lane >> 4;   // 0: lanes 0-15, 1: lanes 16-31
    const int l16  = lane & 15;
    const int bi   = blockIdx.x;  // flattened b*L + i

    // ---- 1. stage pair[b,i,:,:] into LDS (padded), x row, init pads ----
    const float* prow = pair + (size_t)bi * (Ln * CBn);
    for (int idx = tid; idx < (Ln * CBn) / 4; idx += 256) {
        const int j  = idx >> 4;          // 16 float4 per 64-float row
        const int c4 = (idx & 15) << 2;
        const float4 v = reinterpret_cast<const float4*>(prow)[idx];
        float* dst = &P[j * PSTR + c4];
        dst[0] = v.x; dst[1] = v.y; dst[2] = v.z; dst[3] = v.w;
    }
    const float* xr = x + (size_t)bi * Dn;
    xrow[tid]       = xr[tid];
    xrow[tid + 256] = xr[tid + 256];
    if (tid < 16) rsum[tid] = 0.0f;
    // zero qk^T pad columns h = 8..15 (64 rows x 8 = 512 entries)
    {
        const int c2 = tid >> 3, h2 = tid & 7;
        qkT[c2 * 16 + 8 + h2]        = 0.0f;
        qkT[(c2 + 32) * 16 + 8 + h2] = 0.0f;
    }
    __syncthreads();

    // ---- 2. Q row: qrow = xrow @ Wq + bq (2 columns per thread) ----
    {
        float a0 = bq[tid], a1 = bq[tid + 256];
        for (int e = 0; e < Dn; ++e) {
            const float xv = xrow[e];
            a0 += xv * Wq[e * Dn + tid];
            a1 += xv * Wq[e * Dn + tid + 256];
        }
        qrow[tid] = a0; qrow[tid + 256] = a1;
    }
    __syncthreads();

    // ---- 3. qk^T[c][h] = (1/sqrt(HD)) * sum_d qrow[h*64+d] * Wk[c, h*64+d] ----
    {
        const int h = tid >> 6, c = tid & 63;
        const float* wkp = Wk + c * Dn + h * HDn;
        const float* qp  = qrow + h * HDn;
        float acc = 0.0f;
        #pragma unroll 8
        for (int d = 0; d < HDn; ++d) acc += qp[d] * wkp[d];
        qkT[c * 16 + h] = acc * 0.125f;   // 1/sqrt(64)
    }
    __syncthreads();

    // ---- 4. scores GEMM: s(256x16) = P(256x64) @ qkT(64x16) via WMMA f32 ----
    for (int t = 0; t < 2; ++t) {
        const int m0 = (wave + t * 8) * 16;
        v8f acc = {};
        for (int kk = 0; kk < 16; ++kk) {
            const int k = kk * 4 + half * 2;   // K rows {k,k+1} (lo half) / {k+2,k+3} (hi)
            v2f a, b;
            a.x = P[(m0 + l16) * PSTR + k];
            a.y = P[(m0 + l16) * PSTR + k + 1];
            b.x = qkT[k * 16 + l16];
            b.y = qkT[(k + 1) * 16 + l16];
            acc = __builtin_amdgcn_wmma_f32_16x16x4_f32(false, a, false, b,
                                                        (short)0, acc, false, false);
        }
        #pragma unroll
        for (int r = 0; r < 8; ++r)
            s[(m0 + r + half * 8) * SSTR + l16] = acc[r];
    }
    __syncthreads();

    // ---- 5. softmax over j per head (wave w -> head w) ----
    {
        const int h = wave;
        float vals[8];
        float m = -3.4e38f;
        #pragma unroll
        for (int r = 0; r < 8; ++r) {
            vals[r] = s[(lane + r * 32) * SSTR + h];
            m = fmaxf(m, vals[r]);
        }
        for (int off = 16; off > 0; off >>= 1) m = fmaxf(m, __shfl_xor(m, off, 32));
        float sum = 0.0f;
        #pragma unroll
        for (int r = 0; r < 8; ++r) {
            const float e = __expf(vals[r] - m);
            s[(lane + r * 32) * SSTR + h] = e;
            sum += e;
        }
        for (int off = 16; off > 0; off >>= 1) sum += __shfl_xor(sum, off, 32);
        if (lane == 0) rsum[h] = 1.0f / sum;
    }
    __syncthreads();

    // ---- 6. wp GEMM: wp(16x64) = w(16x256) @ P(256x64) via WMMA f32 ----
    if (wave < 4) {
        const int c0 = wave * 16;
        v8f acc = {};
        for (int kk = 0; kk < 64; ++kk) {
            const int k = kk * 4 + half * 2;
            v2f a, b;
            a.x = s[k * SSTR + l16];           // A: rows h=l16, cols j=k..
            a.y = s[(k + 1) * SSTR + l16];
            b.x = P[k * PSTR + c0 + l16];      // B: rows j=k.., cols c
            b.y = P[(k + 1) * PSTR + c0 + l16];
            acc = __builtin_amdgcn_wmma_f32_16x16x4_f32(false, a, false, b,
                                                        (short)0, acc, false, false);
        }
        #pragma unroll
        for (int r = 0; r < 8; ++r) {
            const int h = r + half * 8;
            wp[h * CBn + c0 + l16] = acc[r] * rsum[h];   // normalize here
        }
    }
    __syncthreads();

    // ---- 7. context[h*64+d] = sum_c wp[h][c] * Wv[c, h*64+d] + bv ----
    for (int pass = 0; pass < 2; ++pass) {
        const int dd = tid + pass * 256;
        const int h  = dd >> 6;
        const float* wpr = wp + h * CBn;
        float acc = bv[dd];
        #pragma unroll 8
        for (int c = 0; c < CBn; ++c) acc += wpr[c] * Wv[c * Dn + dd];
        ctx[dd] = acc;
    }
    __syncthreads();

    // ---- 8. out = ctx @ Wo + bo + residual; LayerNorm ----
    float y0, y1;
    {
        float a0 = bo[tid], a1 = bo[tid + 256];
        for (int e = 0; e < Dn; ++e) {
            const float cv = ctx[e];
            a0 += cv * Wo[e * Dn + tid];
            a1 += cv * Wo[e * Dn + tid + 256];
        }
        y0 = a0 + xrow[tid];
        y1 = a1 + xrow[tid + 256];
    }
    float sum = y0 + y1, sq = y0 * y0 + y1 * y1;
    for (int off = 16; off > 0; off >>= 1) {
        sum += __shfl_xor(sum, off, 32);
        sq  += __shfl_xor(sq,  off, 32);
    }
    if (lane == 0) { redS[wave] = sum; redQ[wave] = sq; }
    __syncthreads();
    float tS = 0.0f, tQ = 0.0f;
    #pragma unroll
    for (int w = 0; w < 8; ++w) { tS += redS[w]; tQ += redQ[w]; }
    const float mean = tS * (1.0f / 512.0f);
    const float var  = tQ * (1.0f / 512.0f) - mean * mean;
    const float inv  = rsqrtf(var + 1e-5f);
    float* orow = out + (size_t)bi * Dn;
    orow[tid]       = (y0 - mean) * inv * gamma[tid]       + beta[tid];
    orow[tid + 256] = (y1 - mean) * inv * gamma[tid + 256] + beta[tid + 256];
}

extern "C" void kernel_launch(void* const* d_in, const int* in_sizes, int n_in,
                              void* d_out, int out_size, void* d_ws, size_t ws_size,
                              hipStream_t stream) {
    const float* x     = (const float*)d_in[0];
    const float* pair  = (const float*)d_in[1];
    const float* Wq    = (const float*)d_in[2];
    const float* bq    = (const float*)d_in[3];
    const float* Wk    = (const float*)d_in[4];
    // d_in[5] = bk: provably softmax-invariant, unused
    const float* Wv    = (const float*)d_in[6];
    const float* bv    = (const float*)d_in[7];
    const float* Wo    = (const float*)d_in[8];
    const float* bo    = (const float*)d_in[9];
    const float* gamma = (const float*)d_in[10];
    const float* beta  = (const float*)d_in[11];
    float* out = (float*)d_out;

    dim3 grid(Bn * Ln);   // 512 workgroups, one per (b, i)
    dim3 block(256);      // 8 wave32s
    pair_attn_fused<<<grid, block, 0, stream>>>(x, pair, Wq, bq, Wk, Wv, bv,
                                                Wo, bo, gamma, beta, out);
}